// Adapter_80839874445842
// MI455X (gfx1250) — compile-verified
//
#include <hip/hip_runtime.h>

typedef __attribute__((ext_vector_type(2))) float v2f;
typedef __attribute__((ext_vector_type(4))) float v4f;
typedef __attribute__((ext_vector_type(8))) float v8f;

#define BS   8
#define SQ   2048
#define DIN  2048
#define DOUT 2048
#define NSK  8
#define RLO  16
#define EPSV 1e-12f

// D = A(16x4 f32) * B(4x16 f32) + C(16x16 f32), wave32.
__device__ __forceinline__ v8f wmma_f32(v2f a, v2f b, v8f c) {
  return __builtin_amdgcn_wmma_f32_16x16x4_f32(false, a, false, b, (short)0, c,
                                               false, false);
}

// ---------------------------------------------------------------------------
// Kernel 1: selector probs + weighted-sum of skills -> per-example A_r / B_r
//   A_r[b][i][r] = sum_t p[b,t] * A[t][i][r]
//   B_r[b][r][o] = sum_t p[b,t] * B[t][r][o]
// ---------------------------------------------------------------------------
__global__ __launch_bounds__(256) void combine_kernel(
    const float* __restrict__ logits,   // [32, NSK]
    const int*   __restrict__ tasks,    // [BS]
    const float* __restrict__ A,        // [NSK, DIN, RLO]
    const float* __restrict__ Bm,       // [NSK, RLO, DOUT]
    float* __restrict__ Ar,             // [BS, DIN, RLO]
    float* __restrict__ Br) {           // [BS, RLO, DOUT]
  const int nA = BS * DIN * RLO;
  const int nB = BS * RLO * DOUT;
  int gid = blockIdx.x * blockDim.x + threadIdx.x;
  if (gid >= nA + nB) return;

  int b, rem, stride;
  const float* src;
  float* dst;
  if (gid < nA) {
    stride = DIN * RLO; b = gid / stride; rem = gid % stride; src = A;  dst = Ar;
  } else {
    int g = gid - nA;
    stride = RLO * DOUT; b = g / stride; rem = g % stride; src = Bm; dst = Br;
  }
  const int task = tasks[b];
  float p[NSK];
  float s = 0.f;
#pragma unroll
  for (int t = 0; t < NSK; ++t) {
    p[t] = 1.f / (1.f + __expf(-logits[task * NSK + t]));
    s += p[t];
  }
  const float inv = 1.f / (s + EPSV);
  float acc = 0.f;
#pragma unroll
  for (int t = 0; t < NSK; ++t) acc += p[t] * inv * src[t * stride + rem];
  dst[b * stride + rem] = acc;
}

// ---------------------------------------------------------------------------
// Kernel 2: t[b][s][r] = (1/RLO) * sum_i x[b,s,i] * A_r[b][i][r]
// Workgroup: 256 threads / 8 waves, 128 rows; wave = 16 rows x 16 cols tile.
// ---------------------------------------------------------------------------
__global__ __launch_bounds__(256) void xa_kernel(
    const float* __restrict__ x,        // [BS*SQ, DIN]
    const float* __restrict__ Ar,       // [BS, DIN, RLO]
    float* __restrict__ t) {            // [BS*SQ, RLO]
  __shared__ float Xs[128 * 36];        // padded row stride (bank-friendly)
  __shared__ v2f   ArP[16 * 16];        // K-pairs: {A_r[2kp][n], A_r[2kp+1][n]}

  const int tid  = threadIdx.x;
  const int w    = tid >> 5;
  const int lane = tid & 31;
  const int hi   = lane >> 4;
  const int n16  = lane & 15;
  const int m0   = blockIdx.x * 128;
  const float* Arb = Ar + (m0 / SQ) * (DIN * RLO);

  v8f acc = {};
  for (int kb = 0; kb < DIN; kb += 32) {
#pragma unroll
    for (int j = 0; j < 4; ++j) {                 // 1024 float4 = 128 x 32
      int idx = tid + j * 256;
      int row = idx >> 3, c4 = (idx & 7) << 2;
      *(v4f*)&Xs[row * 36 + c4] = *(const v4f*)&x[(m0 + row) * DIN + kb + c4];
    }
    {
      int kp = tid >> 4, n = tid & 15;            // 256 pairs
      v2f v;
      v.x = Arb[(kb + 2 * kp) * RLO + n];
      v.y = Arb[(kb + 2 * kp + 1) * RLO + n];
      ArP[kp * 16 + n] = v;
    }
    __syncthreads();
#pragma unroll
    for (int k0 = 0; k0 < 32; k0 += 4) {
      v2f a  = *(const v2f*)&Xs[(w * 16 + n16) * 36 + k0 + 2 * hi];
      v2f bf = ArP[(k0 / 2 + hi) * 16 + n16];
      acc = wmma_f32(a, bf, acc);
    }
    __syncthreads();
  }
  const float scale = 1.0f / (float)RLO;
#pragma unroll
  for (int v = 0; v < 8; ++v) {
    int row = m0 + w * 16 + v + 8 * hi;           // C layout: VGPR v -> M=v / v+8
    t[row * RLO + n16] = acc[v] * scale;
  }
}

// ---------------------------------------------------------------------------
// Kernel 3: out = x*W^T + bias + t*B_r   (LoRA folded in as 16-deep K-tail)
// Workgroup: 64x128 tile, 8 waves; wave = 32x32 (2x2 WMMA accumulators).
// ---------------------------------------------------------------------------
__global__ __launch_bounds__(256) void gemm_kernel(
    const float* __restrict__ x,        // [BS*SQ, DIN]
    const float* __restrict__ W,        // [DOUT, DIN]
    const float* __restrict__ bias,     // [DOUT]
    const float* __restrict__ t,        // [BS*SQ, RLO]  (pre-scaled by 1/RLO)
    const float* __restrict__ Br,       // [BS, RLO, DOUT]
    float* __restrict__ out) {          // [BS*SQ, DOUT]
  __shared__ float Xs[64 * 36];
  __shared__ v2f   WsP[16 * 129];       // K-pairs of W^T tile (padded stride)

  const int tid  = threadIdx.x;
  const int w    = tid >> 5;
  const int lane = tid & 31;
  const int hi   = lane >> 4;
  const int n16  = lane & 15;
  const int wm   = w & 1;               // 2 waves along M
  const int wn   = w >> 1;              // 4 waves along N
  const int m0   = blockIdx.y * 64;
  const int o0   = blockIdx.x * 128;
  const int mbase = m0 + wm * 32;
  const int nbase = o0 + wn * 32;

  v8f acc[2][2] = {};

  for (int kb = 0; kb < DIN; kb += 32) {
#pragma unroll
    for (int j = 0; j < 2; ++j) {                 // 512 float4 = 64 x 32
      int idx = tid + j * 256;
      int row = idx >> 3, c4 = (idx & 7) << 2;
      *(v4f*)&Xs[row * 36 + c4] = *(const v4f*)&x[(m0 + row) * DIN + kb + c4];
    }
#pragma unroll
    for (int j = 0; j < 8; ++j) {                 // 2048 v2f = 128 rows x 16 kp
      int idx = tid + j * 256;
      int n = idx >> 4, kp = idx & 15;
      WsP[kp * 129 + n] = *(const v2f*)&W[(o0 + n) * DIN + kb + 2 * kp];
    }
    __syncthreads();
#pragma unroll
    for (int k0 = 0; k0 < 32; k0 += 4) {
      v2f a0 = *(const v2f*)&Xs[(wm * 32 + n16) * 36 + k0 + 2 * hi];
      v2f a1 = *(const v2f*)&Xs[(wm * 32 + 16 + n16) * 36 + k0 + 2 * hi];
      v2f b0 = WsP[(k0 / 2 + hi) * 129 + wn * 32 + n16];
      v2f b1 = WsP[(k0 / 2 + hi) * 129 + wn * 32 + 16 + n16];
      acc[0][0] = wmma_f32(a0, b0, acc[0][0]);
      acc[0][1] = wmma_f32(a0, b1, acc[0][1]);
      acc[1][0] = wmma_f32(a1, b0, acc[1][0]);
      acc[1][1] = wmma_f32(a1, b1, acc[1][1]);
    }
    __syncthreads();
  }

  // ---- LoRA K-tail: 16 extra K with A <- t, B <- B_r[b] ----
  const float* Brb = Br + (m0 / SQ) * (RLO * DOUT);
#pragma unroll
  for (int j = 0; j < 4; ++j) {                   // 1024 v2f = 8 kp x 128 n
    int idx = tid + j * 256;
    int kp = idx >> 7, n = idx & 127;
    v2f v;
    v.x = Brb[(2 * kp) * DOUT + o0 + n];
    v.y = Brb[(2 * kp + 1) * DOUT + o0 + n];
    WsP[kp * 129 + n] = v;
  }
  __syncthreads();
#pragma unroll
  for (int k0 = 0; k0 < 16; k0 += 4) {
    v2f a0 = *(const v2f*)&t[(mbase + n16) * RLO + k0 + 2 * hi];
    v2f a1 = *(const v2f*)&t[(mbase + 16 + n16) * RLO + k0 + 2 * hi];
    v2f b0 = WsP[(k0 / 2 + hi) * 129 + wn * 32 + n16];
    v2f b1 = WsP[(k0 / 2 + hi) * 129 + wn * 32 + 16 + n16];
    acc[0][0] = wmma_f32(a0, b0, acc[0][0]);
    acc[0][1] = wmma_f32(a0, b1, acc[0][1]);
    acc[1][0] = wmma_f32(a1, b0, acc[1][0]);
    acc[1][1] = wmma_f32(a1, b1, acc[1][1]);
  }

  const float bv0 = bias[nbase + n16];
  const float bv1 = bias[nbase + 16 + n16];
#pragma unroll
  for (int sm = 0; sm < 2; ++sm) {
#pragma unroll
    for (int v = 0; v < 8; ++v) {
      int row = mbase + sm * 16 + v + 8 * hi;
      out[row * DOUT + nbase + n16]      = acc[sm][0][v] + bv0;
      out[row * DOUT + nbase + 16 + n16] = acc[sm][1][v] + bv1;
    }
  }
}

// ---------------------------------------------------------------------------
extern "C" void kernel_launch(void* const* d_in, const int* in_sizes, int n_in,
                              void* d_out, int out_size, void* d_ws,
                              size_t ws_size, hipStream_t stream) {
  (void)in_sizes; (void)n_in; (void)out_size; (void)ws_size;
  const float* x      = (const float*)d_in[0];
  const float* W      = (const float*)d_in[1];
  const float* bias   = (const float*)d_in[2];
  const float* logits = (const float*)d_in[3];
  const float* A      = (const float*)d_in[4];
  const float* Bm     = (const float*)d_in[5];
  const int*   tasks  = (const int*)d_in[6];
  float* out = (float*)d_out;

  float* Ar = (float*)d_ws;                    // [BS, DIN, RLO]  1 MB
  float* Br = Ar + BS * DIN * RLO;             // [BS, RLO, DOUT] 1 MB
  float* t  = Br + BS * RLO * DOUT;            // [BS*SQ, RLO]    1 MB

  const int nComb = BS * DIN * RLO + BS * RLO * DOUT;
  combine_kernel<<<(nComb + 255) / 256, 256, 0, stream>>>(logits, tasks, A, Bm,
                                                          Ar, Br);
  xa_kernel<<<(BS * SQ) / 128, 256, 0, stream>>>(x, Ar, t);
  dim3 grid(DOUT / 128, (BS * SQ) / 64);
  gemm_kernel<<<grid, 256, 0, stream>>>(x, W, bias, t, Br, out);
}